// PGA_Conv_10900626997255
// MI455X (gfx1250) — compile-verified
//
#include <hip/hip_runtime.h>
#include <math.h>

// PGA hybrid beamforming (K=4 users, B=512 batch, NT=64, M=NRF=8) for gfx1250.
// One block per batch sample b; 8 wave32 per block. 64-wide complex matmuls use
// V_WMMA_F32_16X16X4_F32 (4 real WMMAs per complex k-step; F32 WMMA has no A/B
// NEG per ISA so -Ai is formed in VALU). R tiles (16KB each) are staged into
// LDS by the Tensor Data Mover (tensor_load_to_lds, TENSORcnt) overlapped with
// the 8x8 scalar algebra; WMMA then reads R from LDS. F/W state kept in the
// F/W regions of d_out (block-local, rewritten every call -> deterministic).

#define KU 4
#define BB 512
#define NTT 64
#define MM 8
#define NITER_MAX 10
#define SIGMA2f 1.0f
#define PTf 1.0f
#define LOG2f_ 0.69314718055994530942f
#define DEN_EPSf 1e-4f
#define PROJ_EPSf 1e-12f

typedef __attribute__((ext_vector_type(2))) float v2f;
typedef __attribute__((ext_vector_type(8))) float v8f;
typedef __attribute__((ext_vector_type(4))) unsigned int u32x4;
typedef __attribute__((ext_vector_type(8))) int i32x8;
typedef __attribute__((ext_vector_type(4))) int i32x4;

__device__ __forceinline__ float2 cmul(float2 a, float2 b){ return make_float2(a.x*b.x - a.y*b.y, a.x*b.y + a.y*b.x); }
__device__ __forceinline__ float2 cmulc(float2 a, float2 b){ return make_float2(a.x*b.x + a.y*b.y, a.x*b.y - a.y*b.x); } // conj(a)*b
__device__ __forceinline__ float2 cadd(float2 a, float2 b){ return make_float2(a.x + b.x, a.y + b.y); }
__device__ __forceinline__ float2 csub(float2 a, float2 b){ return make_float2(a.x - b.x, a.y - b.y); }
__device__ __forceinline__ float2 cscale(float2 a, float s){ return make_float2(a.x*s, a.y*s); }
__device__ __forceinline__ float2 cdiv(float2 a, float2 b){
  float inv = 1.0f / (b.x*b.x + b.y*b.y);
  return make_float2((a.x*b.x + a.y*b.y)*inv, (a.y*b.x - a.x*b.y)*inv);
}

// one complex k-step (K=4) of D += A*B on a 16x16 tile: 4 real WMMAs.
__device__ __forceinline__ void cwmma4(v2f aR, v2f aI, v2f bR, v2f bI, v8f& cR, v8f& cI){
  v2f aIn; aIn[0] = -aI[0]; aIn[1] = -aI[1];
  cR = __builtin_amdgcn_wmma_f32_16x16x4_f32(false, aR,  false, bR, (short)0, cR, false, false);
  cR = __builtin_amdgcn_wmma_f32_16x16x4_f32(false, aIn, false, bI, (short)0, cR, false, false);
  cI = __builtin_amdgcn_wmma_f32_16x16x4_f32(false, aR,  false, bI, (short)0, cI, false, false);
  cI = __builtin_amdgcn_wmma_f32_16x16x4_f32(false, aI,  false, bR, (short)0, cI, false, false);
}

// TDM: async-copy one contiguous 16KB (4096 f32) tile from global into LDS.
// D# per CDNA5 ISA ch.8: group0 = {count=1, lds_addr, global_addr, type=2};
// group1 = {data_size=4B, tensor_dim0=4096, tensor_dim1=1, tile_dim0=4096,
// tile_dim1=1, stride0=4096}; groups 2/3 zero (<=2D tensor). Tracked by
// TENSORcnt; EXEC ignored (issue from one wave only).
// This toolchain's builtin is the 6-arg form:
//   (u32x4 g0, i32x8 g1, i32x4 g2, i32x4 g3, i32x8 pad, i32 cpol)
__device__ __forceinline__ void tdm_load_16kb(const float* gsrc, unsigned lds_off){
  unsigned long long ga = (unsigned long long)(const void*)gsrc;
  u32x4 g0;
  g0[0] = 1u;                                                   // count=1, user mode
  g0[1] = lds_off;                                              // lds_addr (bytes)
  g0[2] = (unsigned)(ga & 0xFFFFFFFFull);                       // global_addr[31:0]
  g0[3] = (unsigned)((ga >> 32) & 0x01FFFFFFull) | (2u << 30);  // addr[56:32] | type=2
  i32x8 g1;
  g1[0] = (int)(2u << 16);        // workgroup_mask=0, data_size=2 (4 bytes)
  g1[1] = (int)(4096u << 16);     // tensor_dim0[15:0]=4096 in [31:16]
  g1[2] = (int)(1u << 16);        // tensor_dim0 hi=0 | tensor_dim1[15:0]=1
  g1[3] = (int)(4096u << 16);     // tensor_dim1 hi=0 | tile_dim0=4096
  g1[4] = 1;                      // tile_dim1=1, tile_dim2=0
  g1[5] = 4096;                   // tensor_dim0_stride[31:0]
  g1[6] = 0;                      // stride0 hi | tensor_dim1_stride lo
  g1[7] = 0;
  i32x4 z4; z4[0]=0; z4[1]=0; z4[2]=0; z4[3]=0;
  i32x8 z8; z8[0]=0; z8[1]=0; z8[2]=0; z8[3]=0; z8[4]=0; z8[5]=0; z8[6]=0; z8[7]=0;
  __builtin_amdgcn_tensor_load_to_lds(g0, g1, z4, z4, z8, 0);
}

extern "C" __global__ void __launch_bounds__(256)
pga_kernel(const float* __restrict__ Hr, const float* __restrict__ Hi,
           const float* __restrict__ Rr, const float* __restrict__ Ri,
           const float* __restrict__ F0p, const float* __restrict__ W0r,
           const float* __restrict__ W0i, const float* __restrict__ step,
           const int* __restrict__ nit_p, float* __restrict__ out)
{
  const int b    = blockIdx.x;
  const int tid  = threadIdx.x;
  const int lane = tid & 31;
  const int wave = tid >> 5;

  int n = nit_p[0];
  if (n < 0) n = 0;
  if (n > NITER_MAX) n = NITER_MAX;

  float*  rates = out;                                   // [B][n+1]
  float*  taus  = out + (size_t)BB*(n+1);                // [B][n+1]
  float2* Fst   = (float2*)(out + (size_t)2*BB*(n+1));   // [K][B][NT][M] complex
  float2* Wst   = Fst + (size_t)KU*BB*NTT*MM;            // [K][B][M][M]  complex

  __shared__ float2 sE[NTT][NTT];   // 32KB: E / E2 / E3
  __shared__ float  sRr[NTT*NTT];   // 16KB: TDM-staged Rr tile
  __shared__ float  sRi[NTT*NTT];   // 16KB: TDM-staged Ri tile
  __shared__ float2 sF[NTT][MM];    // current-k F
  __shared__ float2 sH[NTT][MM];    // current-k H
  __shared__ float2 sA[NTT][MM];    // F@V (or FV)
  __shared__ float2 sX8[NTT][MM];   // F@W
  __shared__ float2 sT[NTT][MM];    // E @ X result
  __shared__ float2 sG[NTT][MM];    // gF accumulator (over k)
  __shared__ float2 sU[MM][MM];     // u = F^H H (as u[m][r])
  __shared__ float2 sV[MM][MM];     // W W^H
  __shared__ float2 sV1[MM][MM];    // conj(u) V
  __shared__ float2 sQ[MM][MM];     // u^H W
  __shared__ float2 sW[MM][MM];
  __shared__ float2 sGW[MM][MM];    // gW_com
  __shared__ float2 sXm[MM][MM];    // v1/d1 - v2/d2
  __shared__ float2 sp8[MM], sd1[MM], sd2[MM];
  __shared__ float  sRed[3];        // 0: rate, 1: tau, 2: pw

  // ---------------- small helpers (block-uniform control flow) ----------------
  auto loadF = [&](int k){
    size_t base = ((size_t)k*BB + b)*NTT*MM;
    for (int idx = tid; idx < NTT*MM; idx += 256) sF[idx>>3][idx&7] = Fst[base + idx];
  };
  auto loadW = [&](int k){
    if (tid < MM*MM) sW[tid>>3][tid&7] = Wst[((size_t)k*BB + b)*MM*MM + tid];
  };
  auto loadH = [&](int k){
    size_t base = ((size_t)k*BB + b)*NTT*MM;
    for (int idx = tid; idx < NTT*MM; idx += 256)
      sH[idx>>3][idx&7] = make_float2(Hr[base+idx], Hi[base+idx]);
  };
  auto issueR = [&](int k){          // async: stage R_k into LDS via TDM
    if (wave == 0){
      size_t rbase = ((size_t)k*BB + b)*NTT*NTT;
      tdm_load_16kb(Rr + rbase, (unsigned)(unsigned long long)(void*)&sRr[0]);
      tdm_load_16kb(Ri + rbase, (unsigned)(unsigned long long)(void*)&sRi[0]);
    }
  };
  auto waitR = [&](){                // completion fence before WMMA consumes R
    if (wave == 0) __builtin_amdgcn_s_wait_tensorcnt(0);
  };

  auto uv_phase = [&](){            // sU = F^H H ; sV = W W^H
    if (tid < 64) {
      int m = tid >> 3, r = tid & 7;
      float2 acc = make_float2(0.f, 0.f);
      for (int t = 0; t < NTT; ++t) acc = cadd(acc, cmulc(sF[t][r], sH[t][m]));
      sU[m][r] = acc;
    } else if (tid < 128) {
      int e = tid - 64, r = e >> 3, s = e & 7;
      float2 acc = make_float2(0.f, 0.f);
      for (int q = 0; q < MM; ++q)
        acc = cadd(acc, make_float2(sW[r][q].x*sW[s][q].x + sW[r][q].y*sW[s][q].y,
                                    sW[r][q].y*sW[s][q].x - sW[r][q].x*sW[s][q].y));
      sV[r][s] = acc;
    }
  };

  auto v1p_phase = [&](){           // sV1 = conj(u) V ; sp8 = u^H w
    if (tid < 64) {
      int m = tid >> 3, s = tid & 7;
      float2 acc = make_float2(0.f, 0.f);
      for (int r = 0; r < MM; ++r) acc = cadd(acc, cmulc(sU[m][r], sV[r][s]));
      sV1[m][s] = acc;
    } else if (tid < 72) {
      int m = tid - 64;
      float2 acc = make_float2(0.f, 0.f);
      for (int r = 0; r < MM; ++r) acc = cadd(acc, cmulc(sU[m][r], sW[r][m]));
      sp8[m] = acc;
    }
  };

  auto tr_phase = [&](){            // d1/d2 from tr, tr1 (complex, per reference)
    if (tid < MM) {
      int m = tid;
      float2 acc = make_float2(0.f, 0.f);
      for (int s = 0; s < MM; ++s) acc = cadd(acc, cmul(sV1[m][s], sU[m][s]));
      float pp = sp8[m].x*sp8[m].x + sp8[m].y*sp8[m].y;
      sd1[m] = make_float2(LOG2f_*(acc.x + SIGMA2f) + DEN_EPSf, LOG2f_*acc.y);
      sd2[m] = make_float2(LOG2f_*(acc.x - pp + SIGMA2f) + DEN_EPSf, LOG2f_*acc.y);
    }
  };

  auto fv_phase = [&](){            // sA = F @ V
    for (int idx = tid; idx < NTT*MM; idx += 256){
      int t = idx >> 3, s = idx & 7;
      float2 acc = make_float2(0.f, 0.f);
      for (int r = 0; r < MM; ++r) acc = cadd(acc, cmul(sF[t][r], sV[r][s]));
      sA[t][s] = acc;
    }
  };

  auto x_phase = [&](){             // sXm = v1/d1 - v2/d2
    if (tid < 64){
      int m = tid >> 3, s = tid & 7;
      float2 cw = make_float2(sW[s][m].x, -sW[s][m].y);      // conj(w[m][s])
      float2 v2 = csub(sV1[m][s], cmul(sp8[m], cw));
      sXm[m][s] = csub(cdiv(sV1[m][s], sd1[m]), cdiv(v2, sd2[m]));
    }
  };

  // sE = sA(64x8) @ F^H(8x64) - R_k(LDS)  (16 tiles of 16x16, 2 tiles per wave)
  auto wmmaE = [&](){
    int half = lane >> 4, l15 = lane & 15;
    for (int tile = wave; tile < 16; tile += 8){
      int ti = tile >> 2, tj = tile & 3;
      v8f cR = {}; v8f cI = {};
      for (int s2 = 0; s2 < 2; ++s2){
        int k0  = 4*s2 + 2*half;
        int row = ti*16 + l15;
        int col = tj*16 + l15;
        float2 a0 = sA[row][k0], a1 = sA[row][k0+1];
        float2 f0 = sF[col][k0], f1 = sF[col][k0+1];
        v2f aR, aI, bR, bI;
        aR[0]=a0.x; aR[1]=a1.x; aI[0]=a0.y; aI[1]=a1.y;
        bR[0]=f0.x; bR[1]=f1.x; bI[0]=-f0.y; bI[1]=-f1.y;   // B = F^H (conj)
        cwmma4(aR, aI, bR, bI, cR, cI);
      }
      for (int v = 0; v < 8; ++v){
        int row = ti*16 + v + 8*half;
        int col = tj*16 + l15;
        int gi = row*NTT + col;
        sE[row][col] = make_float2(cR[v] - sRr[gi], cI[v] - sRi[gi]);
      }
    }
  };

  // sT(64x8) = sE(64x64) @ X(64x8); N padded to 16, waves 0..3
  auto wmmaT = [&](const float2 (*X)[MM]){
    if (wave < 4){
      int half = lane >> 4, l15 = lane & 15;
      int ti = wave;
      v8f cR = {}; v8f cI = {};
      for (int s2 = 0; s2 < 16; ++s2){
        int k0  = 4*s2 + 2*half;
        int row = ti*16 + l15;
        float2 a0 = sE[row][k0], a1 = sE[row][k0+1];
        float2 b0 = (l15 < 8) ? X[k0][l15]   : make_float2(0.f, 0.f);
        float2 b1 = (l15 < 8) ? X[k0+1][l15] : make_float2(0.f, 0.f);
        v2f aR, aI, bR, bI;
        aR[0]=a0.x; aR[1]=a1.x; aI[0]=a0.y; aI[1]=a1.y;
        bR[0]=b0.x; bR[1]=b1.x; bI[0]=b0.y; bI[1]=b1.y;
        cwmma4(aR, aI, bR, bI, cR, cI);
      }
      for (int v = 0; v < 8; ++v){
        int row = ti*16 + v + 8*half;
        if (l15 < 8) sT[row][l15] = make_float2(cR[v], cI[v]);
      }
    }
  };

  auto gfacc_phase = [&](){         // sG += H@Xm - 2*T   (per k)
    for (int idx = tid; idx < NTT*MM; idx += 256){
      int t = idx >> 3, s = idx & 7;
      float2 acc = sG[t][s];
      for (int m = 0; m < MM; ++m) acc = cadd(acc, cmul(sH[t][m], sXm[m][s]));
      acc = csub(acc, cscale(sT[t][s], 2.0f));
      sG[t][s] = acc;
    }
  };

  auto Fupdate = [&](float ss0){    // F += (ss0/K)*sG ; unit-modulus projection
    float sc = ss0 / (float)KU;
    for (int k = 0; k < KU; ++k){
      size_t base = ((size_t)k*BB + b)*NTT*MM;
      for (int idx = tid; idx < NTT*MM; idx += 256){
        float2 f = cadd(Fst[base+idx], cscale(sG[idx>>3][idx&7], sc));
        float mag = sqrtf(f.x*f.x + f.y*f.y);
        f = (mag > PROJ_EPSf) ? cscale(f, 1.0f/mag) : make_float2(0.f, 0.f);
        Fst[base+idx] = f;
      }
    }
  };

  auto b2_phase = [&](){            // sQ = u^H W ; d from tr2
    if (tid < 64){
      int m = tid >> 3, q = tid & 7;
      float2 acc = make_float2(0.f, 0.f);
      for (int r = 0; r < MM; ++r) acc = cadd(acc, cmulc(sU[m][r], sW[r][q]));
      sQ[m][q] = acc;
    } else if (tid < 72){
      int m = tid - 64;
      float2 acc = make_float2(0.f, 0.f);
      for (int r = 0; r < MM; ++r){
        float2 t1 = make_float2(0.f, 0.f);
        for (int s = 0; s < MM; ++s) t1 = cadd(t1, cmul(sV[r][s], sU[m][s]));
        acc = cadd(acc, cmulc(sU[m][r], t1));
      }
      sd1[m] = make_float2(LOG2f_*(acc.x + SIGMA2f) + DEN_EPSf, LOG2f_*acc.y);
    }
  };

  auto b3_phase = [&](){            // sA = F@V ; sX8 = F@W ; sGW (diag only)
    for (int idx = tid; idx < NTT*MM; idx += 256){
      int t = idx >> 3, s = idx & 7;
      float2 a = make_float2(0.f,0.f), w2 = make_float2(0.f,0.f);
      for (int r = 0; r < MM; ++r){
        a  = cadd(a,  cmul(sF[t][r], sV[r][s]));
        w2 = cadd(w2, cmul(sF[t][r], sW[r][s]));
      }
      sA[t][s] = a; sX8[t][s] = w2;
    }
    if (tid < 64){
      int r = tid >> 3, q = tid & 7;
      float2 g = make_float2(0.f, 0.f);
      if (r == q){
        for (int m = 0; m < MM; ++m) g = cadd(g, cdiv(cmul(sU[m][r], sQ[m][q]), sd1[m]));
        g = cscale(g, 1.0f/(float)KU);
      }
      sGW[r][q] = g;
    }
  };

  auto Wupdate = [&](int k, float ssk){   // gW_rad = (2/K) F^H T ; W += ssk*(gW_com-gW_rad)
    if (tid < 64){
      int r = tid >> 3, q = tid & 7;
      float2 g = make_float2(0.f, 0.f);
      for (int t = 0; t < NTT; ++t) g = cadd(g, cmulc(sF[t][r], sT[t][q]));
      g = cscale(g, 2.0f/(float)KU);
      float2 w = cadd(sW[r][q], cscale(csub(sGW[r][q], g), ssk));
      Wst[((size_t)k*BB + b)*MM*MM + tid] = w;
    }
  };

  auto normalize = [&](){           // scale W so sum_k ||F W||^2 == PT
    if (tid == 0) sRed[2] = 0.f;
    __syncthreads();
    for (int k = 0; k < KU; ++k){
      loadF(k); loadW(k);
      __syncthreads();
      float part = 0.f;
      for (int idx = tid; idx < NTT*MM; idx += 256){
        int t = idx >> 3, q = idx & 7;
        float2 acc = make_float2(0.f, 0.f);
        for (int r = 0; r < MM; ++r) acc = cadd(acc, cmul(sF[t][r], sW[r][q]));
        part += acc.x*acc.x + acc.y*acc.y;
      }
      atomicAdd(&sRed[2], part);
      __syncthreads();
    }
    float scale = sqrtf(PTf / (sRed[2] + 1e-12f));
    {
      int k = tid >> 6, e = tid & 63;
      size_t gi = ((size_t)k*BB + b)*MM*MM + e;
      Wst[gi] = cscale(Wst[gi], scale);
    }
    __syncthreads();
  };

  auto rate_phase = [&](){
    if (tid < MM){
      int m = tid;
      float2 acc = make_float2(0.f, 0.f);
      for (int s = 0; s < MM; ++s) acc = cadd(acc, cmul(sV1[m][s], sU[m][s]));
      float pp = sp8[m].x*sp8[m].x + sp8[m].y*sp8[m].y;
      atomicAdd(&sRed[0], log2f(acc.x + SIGMA2f) - log2f(acc.x - pp + SIGMA2f));
    }
  };

  auto metrics = [&](int slot){
    if (tid == 0){ sRed[0] = 0.f; sRed[1] = 0.f; }
    __syncthreads();
    for (int k = 0; k < KU; ++k){
      issueR(k);                     // TDM fetch overlaps scalar algebra below
      loadF(k); loadW(k); loadH(k);
      __syncthreads();
      uv_phase();
      __syncthreads();
      v1p_phase(); fv_phase();
      __syncthreads();
      rate_phase();
      waitR();
      __syncthreads();
      wmmaE();                       // E3 (R from LDS)
      __syncthreads();
      float part = 0.f;
      for (int i = 0; i < 16; ++i){
        int idx = tid*16 + i;
        float2 e = sE[idx>>6][idx&63];
        part += e.x*e.x + e.y*e.y;
      }
      atomicAdd(&sRed[1], part);
      __syncthreads();
    }
    if (tid == 0){
      rates[(size_t)b*(n+1) + slot] = sRed[0];
      taus [(size_t)b*(n+1) + slot] = sRed[1]/(float)KU;
    }
    __syncthreads();
  };

  // ---------------- initialization: F0 = exp(i 2pi phase), W0 ----------------
  {
    const float TWO_PI = 6.28318530717958647692f;
    for (int idx = tid; idx < KU*NTT*MM; idx += 256){
      int k = idx >> 9, rem = idx & 511;
      size_t gi = ((size_t)k*BB + b)*NTT*MM + rem;
      float sn, cs;
      sincosf(F0p[gi]*TWO_PI, &sn, &cs);
      Fst[gi] = make_float2(cs, sn);
    }
    {
      int k = tid >> 6, e = tid & 63;
      size_t gi = ((size_t)k*BB + b)*MM*MM + e;
      Wst[gi] = make_float2(W0r[gi], W0i[gi]);
    }
    __syncthreads();
  }
  normalize();
  metrics(0);

  // ---------------- PGA iterations ----------------
  for (int it = 0; it < n; ++it){
    float ss0 = step[it*(KU+1) + 0];

    for (int idx = tid; idx < NTT*MM; idx += 256) sG[idx>>3][idx&7] = make_float2(0.f, 0.f);
    __syncthreads();

    // ---- Phase A: F gradients (comm + radar), per user k ----
    for (int k = 0; k < KU; ++k){
      issueR(k);                     // TDM fetch overlaps scalar algebra below
      loadF(k); loadW(k); loadH(k);
      __syncthreads();
      uv_phase();              __syncthreads();
      v1p_phase();             __syncthreads();
      tr_phase();              __syncthreads();
      fv_phase(); x_phase();
      waitR();                 __syncthreads();
      wmmaE();                 __syncthreads();   // E = FV F^H - R
      wmmaT(sA);               __syncthreads();   // T = E @ FV
      gfacc_phase();           __syncthreads();
    }
    Fupdate(ss0);
    __syncthreads();

    // ---- Phase B: W gradients with updated F, per user k ----
    for (int k = 0; k < KU; ++k){
      issueR(k);
      loadF(k); loadW(k); loadH(k);
      __syncthreads();
      uv_phase();              __syncthreads();
      b2_phase();              __syncthreads();
      b3_phase();
      waitR();                 __syncthreads();
      wmmaE();                 __syncthreads();   // E2 = F V F^H - R
      wmmaT(sX8);              __syncthreads();   // T = E2 @ (F W)
      Wupdate(k, step[it*(KU+1) + 1 + k]);
      __syncthreads();
    }

    normalize();
    metrics(it + 1);
  }
}

extern "C" void kernel_launch(void* const* d_in, const int* in_sizes, int n_in,
                              void* d_out, int out_size, void* d_ws, size_t ws_size,
                              hipStream_t stream) {
  (void)in_sizes; (void)n_in; (void)out_size; (void)d_ws; (void)ws_size;
  pga_kernel<<<dim3(BB), dim3(256), 0, stream>>>(
      (const float*)d_in[0], (const float*)d_in[1],   // Hr, Hi
      (const float*)d_in[2], (const float*)d_in[3],   // Rr, Ri
      (const float*)d_in[4],                          // F0_phase
      (const float*)d_in[5], (const float*)d_in[6],   // W0r, W0i
      (const float*)d_in[7],                          // step_size
      (const int*)d_in[8],                            // n_iter_outer
      (float*)d_out);
}